// GraphQNN_65481071397022
// MI455X (gfx1250) — compile-verified
//
#include <hip/hip_runtime.h>

// ---------------------------------------------------------------------------
// 22-qubit real statevector simulation, fused with CDNA5 WMMA.
//
// state is REAL f32 (RY gates + real diagonal keep it real). dim = 2^22,
// 16 MB -> L2 resident on MI455X (192 MB L2).
//
// Per layer: 6 Kron-fused 16x16 unitaries applied via chained
// v_wmma_f32_16x16x4_f32 (K=16 = 4 chained k=4 WMMAs), two LDS passes:
//   pass_low : bit positions 0..11  (contiguous 4096-float tiles)
//   pass_high: bit positions 12..21 (strided tiles, sign fused at writeback)
// ---------------------------------------------------------------------------

#define NQ   22
#define DIM  (1 << NQ)            // 4194304
#define NMAT 18                   // 3 layers * 6 group matrices

typedef float v2f __attribute__((ext_vector_type(2)));
typedef float v8f __attribute__((ext_vector_type(8)));

// workspace layout (bytes)
static const size_t STATE_OFF = 0;
static const size_t U_OFF     = (size_t)DIM * 4;          // 16777216
static const size_t RM_OFF    = U_OFF + NMAT * 256 * 4;   // +18432
static const size_t FT_OFF    = RM_OFF + 256;
static const size_t SB_OFF    = FT_OFF + 256;
static const size_t PART_OFF  = SB_OFF + (DIM / 32) * 4;  // +524288

// ---------------------------------------------------------------------------
// prep: row masks for parity, cos/sin table for features, 18 fused 16x16 U's
// ---------------------------------------------------------------------------
__global__ void k_prep(const float* __restrict__ feat,
                       const float* __restrict__ adjm,
                       const float* __restrict__ params,
                       float* __restrict__ U,
                       unsigned* __restrict__ rm,
                       float* __restrict__ ftab) {
  const int tid = threadIdx.x;
  if (tid < NQ) {
    unsigned m = 0;
    for (int j = tid + 1; j < NQ; ++j)
      if (adjm[tid * NQ + j] > 0.0f) m |= 1u << (NQ - 1 - j);
    rm[tid] = m;
    float half = 0.5f * feat[tid];
    ftab[2 * tid + 0] = cosf(half);
    ftab[2 * tid + 1] = sinf(half);
  }
  // groups: global bit-position starts {0,4,8,12,16,18}; group 5 pads with I on p=0,1
  const int S[6] = {0, 4, 8, 12, 16, 18};
  const int i = tid >> 4, j = tid & 15;   // U row (new bits), col (old bits)
  for (int mat = 0; mat < NMAT; ++mat) {
    const int l = mat / 6, g = mat % 6;
    float v = 1.0f;
    for (int p = 0; p < 4; ++p) {
      const int ib = (i >> p) & 1, jb = (j >> p) & 1;
      const bool active = (g < 5) || (p >= 2);
      float e;
      if (active) {
        const int q = NQ - 1 - (S[g] + p);
        const float th = 0.5f * params[l * NQ + q];
        const float c = cosf(th), sn = sinf(th);
        // RY: new0 = c*a - s*b ; new1 = s*a + c*b
        e = ib ? (jb ? c : sn) : (jb ? -sn : c);
      } else {
        e = (ib == jb) ? 1.0f : 0.0f;
      }
      v *= e;
    }
    U[mat * 256 + tid] = v;
  }
}

// ---------------------------------------------------------------------------
// sign bitmask: parity(idx) = XOR_i ( bit_i(idx) & popc(idx & rm[i]) )
// ---------------------------------------------------------------------------
__global__ void k_sign(const unsigned* __restrict__ rm, unsigned* __restrict__ sbits) {
  const int w = blockIdx.x * blockDim.x + threadIdx.x;   // 131072 words
  const unsigned base = (unsigned)w << 5;
  unsigned word = 0;
  for (int b = 0; b < 32; ++b) {
    const unsigned idx = base + b;
    unsigned par = 0;
    #pragma unroll
    for (int i = 0; i < NQ; ++i)
      if ((idx >> (NQ - 1 - i)) & 1u) par ^= (unsigned)__popc(idx & rm[i]);
    word |= (par & 1u) << b;
  }
  sbits[w] = word;
}

// ---------------------------------------------------------------------------
// init: product state after the 22 feature RYs (closed form)
// ---------------------------------------------------------------------------
__global__ void k_init(const float* __restrict__ ftab, float* __restrict__ state) {
  __shared__ float ft[2 * NQ];
  if (threadIdx.x < 2 * NQ) ft[threadIdx.x] = ftab[threadIdx.x];
  __syncthreads();
  const unsigned t = blockIdx.x * blockDim.x + threadIdx.x;  // 1M threads
  const unsigned idx0 = t * 4u;
  float pre = 1.0f;
  #pragma unroll
  for (int p = 2; p < NQ; ++p) {
    const int q = NQ - 1 - p;
    pre *= ft[2 * q + ((idx0 >> p) & 1u)];
  }
  const float c1 = ft[2 * (NQ - 2) + 0], s1 = ft[2 * (NQ - 2) + 1]; // bit pos 1
  const float c0 = ft[2 * (NQ - 1) + 0], s0 = ft[2 * (NQ - 1) + 1]; // bit pos 0
  float4 v;
  v.x = pre * c1 * c0;
  v.y = pre * c1 * s0;
  v.z = pre * s1 * c0;
  v.w = pre * s1 * s0;
  *(float4*)(state + idx0) = v;
}

// ---------------------------------------------------------------------------
// One 16x16 D-tile: D = U(16x16) * B(16x16) via 4 chained wmma_f32_16x16x4.
// tile addresses use the pad-swizzle  addr = reg + x + (x>>4)  (bank-safe).
// Lane roles (wave32): c = lane&15 (column / A-row), h = lane>>4 (half).
// A chunk k: {U[c][4k+2h], U[c][4k+2h+1]}; B chunk k: rows 4k+2h,4k+2h+1 of col c.
// D: VGPR v holds element (m = v + 8h, col c).
// ---------------------------------------------------------------------------
__device__ __forceinline__ void wmma_tile16(float* __restrict__ tile, int regBase,
                                            int x0, int sLoc,
                                            const float* __restrict__ u,
                                            int c, int h) {
  v8f acc = {0.f, 0.f, 0.f, 0.f, 0.f, 0.f, 0.f, 0.f};
  #pragma unroll
  for (int k = 0; k < 4; ++k) {
    const int m0 = 4 * k + 2 * h;
    v2f a, b;
    a.x = u[c * 16 + m0];
    a.y = u[c * 16 + m0 + 1];
    const int xa = x0 + (m0 << sLoc);
    const int xb = x0 + ((m0 + 1) << sLoc);
    b.x = tile[regBase + xa + (xa >> 4)];
    b.y = tile[regBase + xb + (xb >> 4)];
    acc = __builtin_amdgcn_wmma_f32_16x16x4_f32(false, a, false, b,
                                                (short)0, acc, false, false);
  }
  #pragma unroll
  for (int v = 0; v < 8; ++v) {
    const int m = v + 8 * h;
    const int xm = x0 + (m << sLoc);
    tile[regBase + xm + (xm >> 4)] = acc[v];
  }
}

// ---------------------------------------------------------------------------
// pass_low: contiguous 4096-float tiles, groups at bit positions 0,4,8
// ---------------------------------------------------------------------------
__global__ void __launch_bounds__(256) k_pass_low(float* __restrict__ state,
                                                  const float* __restrict__ Umats) {
  __shared__ float tile[4352];   // 4096 + pad
  const int tid = threadIdx.x;
  const unsigned base = blockIdx.x * 4096u;
  #pragma unroll
  for (int i = 0; i < 4; ++i) {
    const int x = (i * 256 + tid) * 4;
    const float4 v = *(const float4*)(state + base + x);
    const int px = x + (x >> 4);
    tile[px + 0] = v.x; tile[px + 1] = v.y; tile[px + 2] = v.z; tile[px + 3] = v.w;
  }
  __syncthreads();
  const int lane = tid & 31, c = lane & 15, h = lane >> 4, wave = tid >> 5; // 8 waves
  const int S[3] = {0, 4, 8};
  for (int g = 0; g < 3; ++g) {
    const float* u = Umats + g * 256;
    const int s = S[g];
    #pragma unroll
    for (int ti = 0; ti < 2; ++ti) {
      const int t = (wave * 2 + ti) * 16 + c;         // 256 columns total
      const int r = t & ((1 << s) - 1);
      const int o = t >> s;
      const int x0 = (o << (s + 4)) + r;
      wmma_tile16(tile, 0, x0, s, u, c, h);
    }
    __syncthreads();
  }
  #pragma unroll
  for (int i = 0; i < 4; ++i) {
    const int x = (i * 256 + tid) * 4;
    const int px = x + (x >> 4);
    const float4 v = make_float4(tile[px], tile[px + 1], tile[px + 2], tile[px + 3]);
    *(float4*)(state + base + x) = v;
  }
}

// ---------------------------------------------------------------------------
// pass_high: strided tiles. Each block owns 8 consecutive low-offsets b and the
// full 1024-point k-space (idx = (k<<12) + b). Groups at k-bit positions 0,4,6
// (= global bits 12-15, 16-19, 18-21 with I-padding). Sign fused at writeback.
// ---------------------------------------------------------------------------
#define NB 8
#define RS 1089   // region stride (1024 + 64 pad + 1 to avoid bank-0 collision)
__global__ void __launch_bounds__(512) k_pass_high(float* __restrict__ state,
                                                   const float* __restrict__ Umats,
                                                   const unsigned* __restrict__ sbits) {
  __shared__ float tile[NB * RS];   // 8712 floats
  const int tid = threadIdx.x;
  const unsigned b0 = blockIdx.x * NB;
  #pragma unroll
  for (int it = 0; it < 16; ++it) {
    const int e = it * 512 + tid;
    const int bi = e & (NB - 1);
    const int k  = e >> 3;
    const unsigned idx = ((unsigned)k << 12) + b0 + bi;
    tile[bi * RS + k + (k >> 4)] = state[idx];
  }
  __syncthreads();
  const int lane = tid & 31, c = lane & 15, h = lane >> 4, wave = tid >> 5; // 16 waves
  const int S[3] = {0, 4, 6};
  for (int g = 0; g < 3; ++g) {
    const float* u = Umats + g * 256;
    const int s = S[g];
    #pragma unroll
    for (int ti = 0; ti < 2; ++ti) {
      const int t  = (wave * 2 + ti) * 16 + c;        // 512 columns total
      const int bi = t >> 6;                          // 64 columns per region
      const int uu = t & 63;
      const int r  = uu & ((1 << s) - 1);
      const int o  = uu >> s;
      const int x0 = (o << (s + 4)) + r;
      wmma_tile16(tile, bi * RS, x0, s, u, c, h);
    }
    __syncthreads();
  }
  #pragma unroll
  for (int it = 0; it < 16; ++it) {
    const int e = it * 512 + tid;
    const int bi = e & (NB - 1);
    const int k  = e >> 3;
    const unsigned idx = ((unsigned)k << 12) + b0 + bi;
    float v = tile[bi * RS + k + (k >> 4)];
    const unsigned w = sbits[idx >> 5];
    if ((w >> (idx & 31u)) & 1u) v = -v;
    state[idx] = v;
  }
}

// ---------------------------------------------------------------------------
// deterministic two-stage expectation-value reduction
// ---------------------------------------------------------------------------
__global__ void __launch_bounds__(256) k_red1(const float* __restrict__ state,
                                              float* __restrict__ part) {
  __shared__ float red[NQ * 256];
  float acc[NQ];
  #pragma unroll
  for (int q = 0; q < NQ; ++q) acc[q] = 0.f;
  const unsigned base = blockIdx.x * 8192u;
  for (int it = 0; it < 32; ++it) {
    const unsigned idx = base + it * 256 + threadIdx.x;
    const float a = state[idx];
    const float p = a * a;
    #pragma unroll
    for (int q = 0; q < NQ; ++q)
      acc[q] += ((idx >> (NQ - 1 - q)) & 1u) ? -p : p;
  }
  #pragma unroll
  for (int q = 0; q < NQ; ++q) red[q * 256 + threadIdx.x] = acc[q];
  __syncthreads();
  for (int off = 128; off > 0; off >>= 1) {
    if (threadIdx.x < (unsigned)off)
      for (int q = 0; q < NQ; ++q)
        red[q * 256 + threadIdx.x] += red[q * 256 + threadIdx.x + off];
    __syncthreads();
  }
  if (threadIdx.x < NQ) part[blockIdx.x * NQ + threadIdx.x] = red[threadIdx.x * 256];
}

__global__ void k_red2(const float* __restrict__ part, float* __restrict__ out) {
  const int q = threadIdx.x;
  if (q < NQ) {
    float s = 0.f;
    for (int b = 0; b < 512; ++b) s += part[b * NQ + q];
    out[q] = s;
  }
}

// ---------------------------------------------------------------------------
extern "C" void kernel_launch(void* const* d_in, const int* in_sizes, int n_in,
                              void* d_out, int out_size, void* d_ws, size_t ws_size,
                              hipStream_t stream) {
  (void)in_sizes; (void)n_in; (void)out_size; (void)ws_size;
  const float* features = (const float*)d_in[0];
  const float* adjm     = (const float*)d_in[1];
  const float* params   = (const float*)d_in[2];
  float* out = (float*)d_out;

  char* ws = (char*)d_ws;
  float*    state = (float*)(ws + STATE_OFF);
  float*    U     = (float*)(ws + U_OFF);
  unsigned* rm    = (unsigned*)(ws + RM_OFF);
  float*    ftab  = (float*)(ws + FT_OFF);
  unsigned* sbits = (unsigned*)(ws + SB_OFF);
  float*    part  = (float*)(ws + PART_OFF);

  k_prep<<<1, 256, 0, stream>>>(features, adjm, params, U, rm, ftab);
  k_sign<<<512, 256, 0, stream>>>(rm, sbits);
  k_init<<<DIM / (256 * 4), 256, 0, stream>>>(ftab, state);
  for (int l = 0; l < 3; ++l) {
    k_pass_low<<<DIM / 4096, 256, 0, stream>>>(state, U + (size_t)(l * 6) * 256);
    k_pass_high<<<4096 / NB, 512, 0, stream>>>(state, U + (size_t)(l * 6 + 3) * 256, sbits);
  }
  k_red1<<<512, 256, 0, stream>>>(state, part);
  k_red2<<<1, 32, 0, stream>>>(part, out);
}